// Suspend_73521250173492
// MI455X (gfx1250) — compile-verified
//
#include <hip/hip_runtime.h>
#include <math.h>

#define VOC   50257
#define OUTD  512
#define HID   64
#define CUR   16
#define BSZ   32
#define SLEN  4096
#define NCH   (SLEN / CUR)   // 256 chunks
#define KCMB  (HID + CUR)    // 80
#define KQ    (KCMB / 4)     // 20 WMMA steps for K=80
#define EPSF  1e-5f

typedef float v2f __attribute__((ext_vector_type(2)));
typedef float v8f __attribute__((ext_vector_type(8)));

// D = A(16x4, f32) * B(4x16, f32) + C(16x16, f32)
__device__ __forceinline__ v8f wmma4(v2f a, v2f b, v8f c) {
  return __builtin_amdgcn_wmma_f32_16x16x4_f32(
      /*neg_a=*/false, a, /*neg_b=*/false, b,
      /*c_mod=*/(short)0, c, /*reuse_a=*/false, /*reuse_b=*/false);
}

// Reductions across the 16-lane halves of a wave32 (lanes 0-15 / 16-31 separate).
__device__ __forceinline__ float half16_sum(float v) {
  v += __shfl_xor(v, 1, 32);
  v += __shfl_xor(v, 2, 32);
  v += __shfl_xor(v, 4, 32);
  v += __shfl_xor(v, 8, 32);
  return v;
}
__device__ __forceinline__ float half16_max(float v) {
  v = fmaxf(v, __shfl_xor(v, 1, 32));
  v = fmaxf(v, __shfl_xor(v, 2, 32));
  v = fmaxf(v, __shfl_xor(v, 4, 32));
  v = fmaxf(v, __shfl_xor(v, 8, 32));
  return v;
}

// ---------------------------------------------------------------------------
// Phase A: per-(chunk,batch) attention block, fully parallel. One wave/block.
// Writes LN(attn) into cols [0,16) of combined[16][80] tile in workspace.
// ---------------------------------------------------------------------------
__global__ __launch_bounds__(32)
void attn_kernel(const int* __restrict__ x, const float* __restrict__ emb,
                 const float* __restrict__ Wq, const float* __restrict__ bq,
                 const float* __restrict__ Wk, const float* __restrict__ bk,
                 const float* __restrict__ Wv, const float* __restrict__ bv,
                 const float* __restrict__ g_lim, const float* __restrict__ b_lim,
                 float* __restrict__ comb_ws) {
  __shared__ float rd[16][16];
  __shared__ float qs[16][16], ks[16][16], vs[16][16], pr[16][16];
  __shared__ int   ids[16];

  const int bid   = blockIdx.x;          // chunk*BSZ + b
  const int chunk = bid / BSZ;
  const int b     = bid % BSZ;
  const int lane  = threadIdx.x;
  const int m     = lane & 15;
  const bool hi   = lane >= 16;
  const int k0    = hi ? 2 : 0;
  const int k1    = hi ? 3 : 1;

  // Hoisted weight fragments + biases (overlap with the embedding gather).
  v2f wf0[4], wf1[4], wf2[4];
  #pragma unroll
  for (int q = 0; q < 4; ++q) {
    wf0[q].x = Wq[(q * 4 + k0) * 16 + m];  wf0[q].y = Wq[(q * 4 + k1) * 16 + m];
    wf1[q].x = Wk[(q * 4 + k0) * 16 + m];  wf1[q].y = Wk[(q * 4 + k1) * 16 + m];
    wf2[q].x = Wv[(q * 4 + k0) * 16 + m];  wf2[q].y = Wv[(q * 4 + k1) * 16 + m];
  }
  const float bqv = bq[m], bkv = bk[m], bvv = bv[m];
  const float g   = g_lim[m], bo = b_lim[m];

  if (lane < 16) ids[lane] = x[(size_t)b * SLEN + chunk * CUR + lane];
  __syncthreads();
  for (int idx = lane; idx < 256; idx += 32) {
    int r = idx >> 4, c = idx & 15;
    rd[r][c] = emb[(size_t)ids[r] * CUR + c];
  }
  __syncthreads();

  // q/k/v = read @ W + b
  {
    v2f af[4];
    #pragma unroll
    for (int q = 0; q < 4; ++q) {
      af[q].x = rd[m][q * 4 + k0];  af[q].y = rd[m][q * 4 + k1];
    }
    v8f aq = {}, ak = {}, av = {};
    #pragma unroll
    for (int q = 0; q < 4; ++q) {
      aq = wmma4(af[q], wf0[q], aq);
      ak = wmma4(af[q], wf1[q], ak);
      av = wmma4(af[q], wf2[q], av);
    }
    #pragma unroll
    for (int r = 0; r < 8; ++r) {
      int row = hi ? r + 8 : r;
      qs[row][m] = aq[r] + bqv;
      ks[row][m] = ak[r] + bkv;
      vs[row][m] = av[r] + bvv;
    }
  }
  __syncthreads();

  // scores = q @ k^T * 1/sqrt(16);  B[k][n] = k[n][kdim]
  v8f sc = {};
  #pragma unroll
  for (int kk = 0; kk < 16; kk += 4) {
    v2f a, bb;
    a.x  = qs[m][kk + k0];  a.y  = qs[m][kk + k1];
    bb.x = ks[m][kk + k0];  bb.y = ks[m][kk + k1];   // n == m
    sc = wmma4(a, bb, sc);
  }
  #pragma unroll
  for (int r = 0; r < 8; ++r) {
    float s  = sc[r] * 0.25f;
    float mx = half16_max(s);
    float e  = expf(s - mx);
    float sm = half16_sum(e);
    pr[hi ? r + 8 : r][m] = e / sm;
  }
  __syncthreads();

  // watch = P @ v
  v8f wt = {};
  #pragma unroll
  for (int kk = 0; kk < 16; kk += 4) {
    v2f a, bb;
    a.x  = pr[m][kk + k0];  a.y  = pr[m][kk + k1];
    bb.x = vs[kk + k0][m];  bb.y = vs[kk + k1][m];
    wt = wmma4(a, bb, wt);
  }

  // LayerNorm over the 16 columns per row, write watch to workspace.
  const size_t base = (size_t)bid * 16 * KCMB;
  #pragma unroll
  for (int r = 0; r < 8; ++r) {
    float v    = wt[r];
    float mean = half16_sum(v) * (1.0f / 16.0f);
    float d    = v - mean;
    float var  = half16_sum(d * d) * (1.0f / 16.0f);
    float rstd = rsqrtf(var + EPSF);
    int row = hi ? r + 8 : r;
    comb_ws[base + row * KCMB + m] = d * rstd * g + bo;
  }
}

// ---------------------------------------------------------------------------
// Phase B: serial sus recurrence. One block per batch (32 blocks, 4 waves).
// Ws fragments / biases / LN affine params hoisted out of the 256-chunk loop:
// the serial path touches global memory only for watch-load + sus-store.
// ---------------------------------------------------------------------------
__global__ __launch_bounds__(128)
void recur_kernel(const float* __restrict__ Wsm, const float* __restrict__ bsv,
                  const float* __restrict__ g_doubt, const float* __restrict__ b_doubt,
                  float* __restrict__ comb_ws) {
  __shared__ float comb[16][KCMB];
  __shared__ float stmp[16][HID];
  __shared__ float part_s[16][8], part_ss[16][8];
  __shared__ float mean_s[16], rstd_s[16];

  const int b    = blockIdx.x;
  const int t    = threadIdx.x;
  const int lane = t & 31;
  const int w    = t >> 5;        // wave 0..3 -> output cols w*16..w*16+15
  const int m    = lane & 15;
  const bool hi  = lane >= 16;
  const int k0   = hi ? 2 : 0;
  const int k1   = hi ? 3 : 1;
  const int n0   = w * 16;

  // Hoisted invariants
  v2f wsfrag[KQ];
  #pragma unroll
  for (int q = 0; q < KQ; ++q) {
    wsfrag[q].x = Wsm[(q * 4 + k0) * HID + n0 + m];
    wsfrag[q].y = Wsm[(q * 4 + k1) * HID + n0 + m];
  }
  const float bias = bsv[n0 + m];
  const float gdv  = g_doubt[t & 63];   // normalize loop always hits col t&63
  const float bdv  = b_doubt[t & 63];

  for (int idx = t; idx < 16 * HID; idx += 128)
    comb[idx / HID][16 + (idx % HID)] = 0.0f;
  __syncthreads();

  for (int chunk = 0; chunk < NCH; ++chunk) {
    const size_t base = ((size_t)chunk * BSZ + b) * 16 * KCMB;
    // pull watch cols for this chunk; push current sus into the ws tile
    for (int idx = t; idx < 256; idx += 128)
      comb[idx >> 4][idx & 15] = comb_ws[base + (idx >> 4) * KCMB + (idx & 15)];
    for (int idx = t; idx < 16 * HID; idx += 128) {
      int r = idx / HID, c = idx % HID;
      comb_ws[base + r * KCMB + 16 + c] = comb[r][16 + c];
    }
    __syncthreads();

    // sus_pre = comb @ Ws + bs  (16 x 80 x 64); batch A-frag LDS reads first.
    v2f af[KQ];
    #pragma unroll
    for (int q = 0; q < KQ; ++q) {
      af[q].x = comb[m][q * 4 + k0];
      af[q].y = comb[m][q * 4 + k1];
    }
    v8f acc = {};
    #pragma unroll
    for (int q = 0; q < KQ; ++q) acc = wmma4(af[q], wsfrag[q], acc);

    #pragma unroll
    for (int r = 0; r < 8; ++r) {
      float xg = acc[r] + bias;
      float gg = 0.5f * xg * (1.0f + erff(xg * 0.70710678118654752f)); // exact gelu
      stmp[hi ? r + 8 : r][n0 + m] = gg;
    }
    __syncthreads();

    // LayerNorm over HID=64 per row
    {
      int row = t >> 3, j = t & 7;
      float s = 0.f, ss = 0.f;
      #pragma unroll
      for (int c = 0; c < 8; ++c) {
        float v = stmp[row][j * 8 + c];
        s += v; ss += v * v;
      }
      part_s[row][j] = s; part_ss[row][j] = ss;
    }
    __syncthreads();
    if (t < 16) {
      float s = 0.f, ss = 0.f;
      #pragma unroll
      for (int j = 0; j < 8; ++j) { s += part_s[t][j]; ss += part_ss[t][j]; }
      float mean = s * (1.0f / HID);
      float var  = ss * (1.0f / HID) - mean * mean;
      mean_s[t] = mean;
      rstd_s[t] = rsqrtf(var + EPSF);
    }
    __syncthreads();
    for (int idx = t; idx < 16 * HID; idx += 128) {
      int r = idx / HID, c = idx % HID;   // c == t&63 every iteration
      comb[r][16 + c] = (stmp[r][c] - mean_s[r]) * rstd_s[r] * gdv + bdv;
    }
    __syncthreads();
  }
}

// ---------------------------------------------------------------------------
// Phase C: out = LN(comb @ Wd + bd) * g_check + b_check. 8192 blocks x 8 waves.
// B fragments batched per tile and double-buffered across the 4 column tiles
// so each tile pays one load-wait instead of 20.
// ---------------------------------------------------------------------------
__global__ __launch_bounds__(256)
void head_kernel(const float* __restrict__ comb_ws, const float* __restrict__ Wd,
                 const float* __restrict__ bd, const float* __restrict__ g_check,
                 const float* __restrict__ b_check, float* __restrict__ out) {
  __shared__ float comb[16][KCMB];
  __shared__ float red_s[16], red_ss[16];
  __shared__ float mean_s[16], rstd_s[16];

  const int bid   = blockIdx.x;      // chunk*BSZ + b
  const int chunk = bid / BSZ;
  const int b     = bid % BSZ;
  const int t     = threadIdx.x;
  const int lane  = t & 31;
  const int w     = t >> 5;          // 8 waves
  const int m     = lane & 15;
  const bool hi   = lane >= 16;
  const int k0    = hi ? 2 : 0;
  const int k1    = hi ? 3 : 1;
  const int nbase = w * 64 + m;      // tile i column = nbase + i*16

  const size_t base = (size_t)bid * 16 * KCMB;
  for (int idx = t; idx < 16 * KCMB; idx += 256)
    comb[idx / KCMB][idx % KCMB] = comb_ws[base + idx];
  if (t < 16) { red_s[t] = 0.f; red_ss[t] = 0.f; }
  __syncthreads();

  // A fragments resident in registers for all column tiles
  v2f afrag[KQ];
  #pragma unroll
  for (int q = 0; q < KQ; ++q) {
    afrag[q].x = comb[m][q * 4 + k0];
    afrag[q].y = comb[m][q * 4 + k1];
  }

  v2f bfrag[2][KQ];
#define LOAD_BTILE(buf, i)                                        \
  _Pragma("unroll")                                               \
  for (int q = 0; q < KQ; ++q) {                                  \
    bfrag[buf][q].x = Wd[(q * 4 + k0) * OUTD + nbase + (i) * 16]; \
    bfrag[buf][q].y = Wd[(q * 4 + k1) * OUTD + nbase + (i) * 16]; \
  }

  v8f acc[4];
  float gch[4], bch[4];
  float ps[8], pss[8];
  #pragma unroll
  for (int r = 0; r < 8; ++r) { ps[r] = 0.f; pss[r] = 0.f; }

  LOAD_BTILE(0, 0);
  #pragma unroll
  for (int i = 0; i < 4; ++i) {
    if (i < 3) { LOAD_BTILE((i + 1) & 1, i + 1); }   // prefetch next tile
    const int n = nbase + i * 16;
    v8f c = {};
    #pragma unroll
    for (int q = 0; q < KQ; ++q) c = wmma4(afrag[q], bfrag[i & 1][q], c);

    const float bias = bd[n];
    #pragma unroll
    for (int r = 0; r < 8; ++r) {
      c[r] += bias;
      ps[r]  += c[r];
      pss[r] += c[r] * c[r];
    }
    acc[i] = c;
    gch[i] = g_check[n];
    bch[i] = b_check[n];
  }
#undef LOAD_BTILE

  #pragma unroll
  for (int r = 0; r < 8; ++r) {
    float s  = half16_sum(ps[r]);
    float ss = half16_sum(pss[r]);
    if (m == 0) {
      int row = hi ? r + 8 : r;
      atomicAdd(&red_s[row], s);
      atomicAdd(&red_ss[row], ss);
    }
  }
  __syncthreads();
  if (t < 16) {
    float mean = red_s[t] * (1.0f / OUTD);
    float var  = red_ss[t] * (1.0f / OUTD) - mean * mean;
    mean_s[t]  = mean;
    rstd_s[t]  = rsqrtf(var + EPSF);
  }
  __syncthreads();

  #pragma unroll
  for (int i = 0; i < 4; ++i) {
    const int n = nbase + i * 16;
    #pragma unroll
    for (int r = 0; r < 8; ++r) {
      int row = hi ? r + 8 : r;
      float vout = (acc[i][r] - mean_s[row]) * rstd_s[row] * gch[i] + bch[i];
      out[((size_t)b * SLEN + (size_t)chunk * CUR + row) * OUTD + n] = vout;
    }
  }
}

// ---------------------------------------------------------------------------
extern "C" void kernel_launch(void* const* d_in, const int* in_sizes, int n_in,
                              void* d_out, int out_size, void* d_ws, size_t ws_size,
                              hipStream_t stream) {
  (void)in_sizes; (void)n_in; (void)out_size; (void)ws_size;
  const int*   x       = (const int*)d_in[0];
  const float* emb     = (const float*)d_in[1];
  const float* Wq      = (const float*)d_in[2];
  const float* bq      = (const float*)d_in[3];
  const float* Wk      = (const float*)d_in[4];
  const float* bk      = (const float*)d_in[5];
  const float* Wv      = (const float*)d_in[6];
  const float* bv      = (const float*)d_in[7];
  const float* g_lim   = (const float*)d_in[8];
  const float* b_lim   = (const float*)d_in[9];
  const float* Wsm     = (const float*)d_in[10];
  const float* bsv     = (const float*)d_in[11];
  const float* g_doubt = (const float*)d_in[12];
  const float* b_doubt = (const float*)d_in[13];
  const float* Wd      = (const float*)d_in[14];
  const float* bd      = (const float*)d_in[15];
  const float* g_check = (const float*)d_in[16];
  const float* b_check = (const float*)d_in[17];
  float* out     = (float*)d_out;
  float* comb_ws = (float*)d_ws;   // [NCH*BSZ][16][80] f32 = ~42 MB

  attn_kernel<<<dim3(NCH * BSZ), 32, 0, stream>>>(
      x, emb, Wq, bq, Wk, bk, Wv, bv, g_lim, b_lim, comb_ws);
  recur_kernel<<<dim3(BSZ), 128, 0, stream>>>(
      Wsm, bsv, g_doubt, b_doubt, comb_ws);
  head_kernel<<<dim3(NCH * BSZ), 256, 0, stream>>>(
      comb_ws, Wd, bd, g_check, b_check, out);
}